// geomm_net_tmi2024_38568806318243
// MI455X (gfx1250) — compile-verified
//
#include <hip/hip_runtime.h>
#include <hip/hip_bf16.h>

// GeoMM-Net TMI2024 head, MI455X (gfx1250, wave32).
//
// Roofline: oct_feats = 235 MB read-once -> ~10us at 23.3 TB/s HBM; everything
// else must hide under that. The argmax slice-gather collapses algebraically
// into lookups of the pooled tensor, so oct_feats is read exactly once.
// The 1.64 GFLOP conv1d GEMM runs on V_WMMA_F32_16X16X4_F32 (full fp32), with
// the K=65536 reduction SPLIT 32 ways so 2048 waves carry the serially-
// dependent accumulation chains. The K loop is structured ci-outer/k-inner so
// BOTH operand streams use immediate-offset vector loads (no per-step 64-bit
// address arithmetic) -- the body is 2 loads + 1 WMMA per step. Partial tiles
// are reduced in fixed order (bit-deterministic, no float atomics).
// w (16.8 MB), xcat (2 MB) and partials (2 MB) are L2-resident (192 MB L2).

typedef __attribute__((ext_vector_type(2))) float v2f;
typedef __attribute__((ext_vector_type(8))) float v8f;

#define BATCH 4
#define MSLICE 112
#define CCH 512
#define MP 128          // padded slice-axis stride for xcat (covers t+k+koff <= 126)
#define KTOT 65536      // 1024 ci * 64 k
#define TOUT 49         // conv1d output length: 112 - 64 + 1
#define NTILE 64        // 4 batch * 4 o-tiles * 4 t-tiles
#define KSPLIT 32       // K chunks per tile
#define KCHUNK (KTOT / KSPLIT)   // 2048 -> 32 ci x 16 wmma steps per wave

__device__ __forceinline__ float sigmoidf_(float x) {
    return 1.0f / (1.0f + __expf(-x));
}

// ---------------------------------------------------------------------------
// Zero the pad columns (m in [112,128)) of xcat[4][1024][128].
__global__ __launch_bounds__(256) void zeropad_kernel(float* __restrict__ xcat) {
    int tid = blockIdx.x * blockDim.x + threadIdx.x;
    if (tid >= BATCH * 1024 * (MP - MSLICE)) return;
    int row = tid >> 4;               // MP-MSLICE == 16
    int m = MSLICE + (tid & 15);
    xcat[row * MP + m] = 0.0f;
}

// ---------------------------------------------------------------------------
// Pooling: one wave per (b,m,c); 256 contiguous floats -> mean + max.
// Writes transposed xcat[b][c][m] = mean, xcat[b][512+c][m] = max.
__global__ __launch_bounds__(256) void pool_kernel(const float* __restrict__ oct,
                                                   float* __restrict__ xcat) {
    int wid = (blockIdx.x * blockDim.x + threadIdx.x) >> 5;
    int lane = threadIdx.x & 31;
    if (wid >= BATCH * MSLICE * CCH) return;
    int c = wid & (CCH - 1);
    int m = (wid >> 9) % MSLICE;
    int b = wid / (MSLICE * CCH);
    const float4* p = (const float4*)(oct + (size_t)wid * 256);
    float4 v0 = p[lane];          // elements [0,128): coalesced 512B per load
    float4 v1 = p[lane + 32];     // elements [128,256)
    float s = v0.x + v0.y + v0.z + v0.w + v1.x + v1.y + v1.z + v1.w;
    float mx = fmaxf(fmaxf(fmaxf(v0.x, v0.y), fmaxf(v0.z, v0.w)),
                     fmaxf(fmaxf(v1.x, v1.y), fmaxf(v1.z, v1.w)));
    #pragma unroll
    for (int off = 16; off; off >>= 1) {
        s += __shfl_xor(s, off, 32);
        mx = fmaxf(mx, __shfl_xor(mx, off, 32));
    }
    if (lane == 0) {
        xcat[((b * 1024) + c) * MP + m] = s * (1.0f / 256.0f);
        xcat[((b * 1024) + 512 + c) * MP + m] = mx;
    }
}

// ---------------------------------------------------------------------------
// feat_o: per (b,c) scan over M=112 slices.
//   feat[512+c]  = max_m fv[b,m,c]
//   feat[1024+c] = fv[b, argmax_m fv_max[b,m,c], c]   (first-occurrence argmax)
__global__ __launch_bounds__(256) void feato_kernel(const float* __restrict__ xcat,
                                                    float* __restrict__ feat) {
    int tid = blockIdx.x * blockDim.x + threadIdx.x;
    if (tid >= BATCH * CCH) return;
    int b = tid >> 9, c = tid & (CCH - 1);
    const float* fv = xcat + ((b * 1024) + c) * MP;
    const float* fm = xcat + ((b * 1024) + 512 + c) * MP;
    float bestAvg = -INFINITY, bestMaxVal = -INFINITY;
    int bestIdx = 0;
    for (int m = 0; m < MSLICE; ++m) {
        float a = fv[m];
        bestAvg = fmaxf(bestAvg, a);
        float x = fm[m];
        if (x > bestMaxVal) { bestMaxVal = x; bestIdx = m; }
    }
    feat[b * 1536 + 512 + c] = bestAvg;
    feat[b * 1536 + 1024 + c] = fv[bestIdx];
}

// ---------------------------------------------------------------------------
// conv1d GEMM, K-split WMMA stage.
// wid = tile * KSPLIT + kc; one wave per (tile, K-chunk). ci-outer/k-inner loop:
// pointer bumps per ci (wp += 64, bp += MP), fully unrolled 16-step inner loop
// with immediate-offset loads. 16x16 f32 partial stored canonically:
//   part[(tile*KSPLIT + kc)*256 + o_local*16 + t_local]
// A (16x4 f32) lane layout: lanes 0-15 hold K=kk,kk+1; lanes 16-31 kk+2,kk+3.
__global__ __launch_bounds__(256) void conv1d_wmma_kernel(const float* __restrict__ w,
                                                          const float* __restrict__ xcat,
                                                          float* __restrict__ part) {
    const int wid = blockIdx.x * 8 + (threadIdx.x >> 5);  // 0..2047
    const int tile = wid / KSPLIT;           // b(4) x otile(4) x ttile(4)
    const int kc = wid % KSPLIT;
    const int b = tile >> 4;
    const int otile = (tile >> 2) & 3;
    const int ttile = tile & 3;
    const int lane = threadIdx.x & 31;
    const int half = lane >> 4;              // 0: lanes 0-15, 1: lanes 16-31
    const int l15 = lane & 15;
    const int koff = half * 2;

    const int kk0 = kc * KCHUNK;             // multiple of 64
    const int tcol = ttile * 16 + l15;       // output position for this lane's B column
    // A row pointer (8B aligned: kk0%4==0, koff even) and B pointer for ci0.
    const float* wp = w + (size_t)(otile * 16 + l15) * KTOT + koff + kk0;
    const float* bp = xcat + (size_t)b * 1024 * MP + (kk0 >> 6) * MP + tcol + koff;

    v8f acc = {};
    for (int ci = 0; ci < KCHUNK / 64; ++ci) {   // 32 channels per chunk
        #pragma unroll
        for (int k = 0; k < 64; k += 4) {        // 16 WMMA steps, immediate offsets
            v2f a = *(const v2f*)(wp + k);       // w[o][ci*64 + k + koff (+1)]
            v2f bv;
            bv.x = bp[k];                        // xcat[b][ci][tcol + k + koff]
            bv.y = bp[k + 1];
            acc = __builtin_amdgcn_wmma_f32_16x16x4_f32(
                /*neg_a=*/false, a, /*neg_b=*/false, bv,
                /*c_mod=*/(short)0, acc, /*reuse_a=*/false, /*reuse_b=*/false);
        }
        wp += 64;
        bp += MP;
    }
    // C/D layout: VGPR j, lanes 0-15 -> row M=j; lanes 16-31 -> row M=j+8; N=lane&15.
    float* pt = part + (size_t)(tile * KSPLIT + kc) * 256;
    #pragma unroll
    for (int j = 0; j < 8; ++j)
        pt[(j + half * 8) * 16 + l15] = acc[j];
}

// ---------------------------------------------------------------------------
// Fixed-order reduction of the KSPLIT partials + bias + sigmoid -> sw[4,64,49].
__global__ __launch_bounds__(256) void conv1d_reduce_kernel(const float* __restrict__ part,
                                                            const float* __restrict__ bias,
                                                            float* __restrict__ sw) {
    int tid = blockIdx.x * blockDim.x + threadIdx.x;   // [0, 4*64*64)
    if (tid >= BATCH * 64 * 64) return;
    int t = tid & 63;
    int o = (tid >> 6) & 63;
    int b = tid >> 12;
    int tile = (b << 4) | ((o >> 4) << 2) | (t >> 4);
    int elem = (o & 15) * 16 + (t & 15);
    const float* pt = part + (size_t)tile * KSPLIT * 256 + elem;
    float s = 0.0f;
    for (int kc = 0; kc < KSPLIT; ++kc)   // fixed order: deterministic
        s += pt[kc * 256];
    if (t < TOUT)
        sw[(b * 64 + o) * TOUT + t] = sigmoidf_(s + bias[o]);
}

// ---------------------------------------------------------------------------
// convf: conv1d(sw[4,64,49], wf[13,64,5], stride 4, pad 2) -> sigmoid -> r[4,13,13]
__global__ __launch_bounds__(128) void convf_kernel(const float* __restrict__ sw,
                                                    const float* __restrict__ wf,
                                                    const float* __restrict__ bf,
                                                    float* __restrict__ r) {
    int tid = blockIdx.x * blockDim.x + threadIdx.x;
    if (tid >= BATCH * 13 * 13) return;
    int p = tid % 13;
    int oc = (tid / 13) % 13;
    int b = tid / 169;
    float acc = bf[oc];
    int base = 4 * p - 2;
    for (int i = 0; i < 64; ++i) {
        const float* srow = sw + (b * 64 + i) * TOUT;
        const float* wrow = wf + (oc * 64 + i) * 5;
        #pragma unroll
        for (int k = 0; k < 5; ++k) {
            int pos = base + k;
            if (pos >= 0 && pos < TOUT) acc += srow[pos] * wrow[k];
        }
    }
    r[b * 169 + oc * 13 + p] = sigmoidf_(acc);
}

// ---------------------------------------------------------------------------
// feat_f[b,c] = mean_hw( features_f[b,c,h,w] * (1 + r[b,h,w]) ); r staged in LDS.
__global__ __launch_bounds__(512) void featf_kernel(const float* __restrict__ ff,
                                                    const float* __restrict__ r,
                                                    float* __restrict__ feat) {
    __shared__ float rs[169];
    int b = blockIdx.x;
    int t = threadIdx.x;
    if (t < 169) rs[t] = 1.0f + r[b * 169 + t];
    __syncthreads();
    const float* p = ff + (size_t)(b * CCH + t) * 169;
    float s = 0.0f;
    for (int j = 0; j < 169; ++j) s += p[j] * rs[j];
    feat[b * 1536 + t] = s * (1.0f / 169.0f);
}

// ---------------------------------------------------------------------------
// Head: out[b,j] = feat[b,:] . head_w[j,:] + head_b[j]; one wave per (b,j).
__global__ __launch_bounds__(384) void head_kernel(const float* __restrict__ feat,
                                                   const float* __restrict__ hw,
                                                   const float* __restrict__ hb,
                                                   float* __restrict__ out) {
    int wave = threadIdx.x >> 5;  // 0..11 -> (b,j)
    int lane = threadIdx.x & 31;
    int b = wave / 3, j = wave % 3;
    float s = 0.0f;
    for (int i = lane; i < 1536; i += 32)
        s += feat[b * 1536 + i] * hw[j * 1536 + i];
    #pragma unroll
    for (int off = 16; off; off >>= 1) s += __shfl_xor(s, off, 32);
    if (lane == 0) out[b * 3 + j] = s + hb[j];
}

// ---------------------------------------------------------------------------
extern "C" void kernel_launch(void* const* d_in, const int* in_sizes, int n_in,
                              void* d_out, int out_size, void* d_ws, size_t ws_size,
                              hipStream_t stream) {
    const float* features_f = (const float*)d_in[0]; // [4,512,13,13]
    const float* oct        = (const float*)d_in[1]; // [4,112,512,16,16]
    const float* w1         = (const float*)d_in[2]; // [64,1024,64]
    const float* b1         = (const float*)d_in[3]; // [64]
    const float* wf         = (const float*)d_in[4]; // [13,64,5]
    const float* bf         = (const float*)d_in[5]; // [13]
    const float* hw         = (const float*)d_in[6]; // [3,1536]
    const float* hb         = (const float*)d_in[7]; // [3]
    float* out = (float*)d_out;                      // [4,3]

    // Workspace layout (floats): xcat | partials | sw | r | feat  (~4.3 MB)
    float* ws   = (float*)d_ws;
    float* xcat = ws;                                  // 4*1024*128       = 524288
    float* part = xcat + BATCH * 1024 * MP;            // 64*32*256        = 524288
    float* sw   = part + (size_t)NTILE * KSPLIT * 256; // 4*64*49          = 12544
    float* rbuf = sw + BATCH * 64 * TOUT;              // 4*13*13          = 676
    float* feat = rbuf + BATCH * 169;                  // 4*1536           = 6144

    zeropad_kernel<<<(BATCH * 1024 * 16 + 255) / 256, 256, 0, stream>>>(xcat);
    pool_kernel<<<(BATCH * MSLICE * CCH * 32) / 256, 256, 0, stream>>>(oct, xcat);
    feato_kernel<<<(BATCH * CCH + 255) / 256, 256, 0, stream>>>(xcat, feat);
    conv1d_wmma_kernel<<<(NTILE * KSPLIT) / 8, 256, 0, stream>>>(w1, xcat, part);
    conv1d_reduce_kernel<<<(BATCH * 64 * 64 + 255) / 256, 256, 0, stream>>>(part, b1, sw);
    convf_kernel<<<(BATCH * 169 + 127) / 128, 128, 0, stream>>>(sw, wf, bf, rbuf);
    featf_kernel<<<BATCH, 512, 0, stream>>>(features_f, rbuf, feat);
    head_kernel<<<1, 384, 0, stream>>>(feat, hw, hb, out);
}